// VFE_Block_19198503813593
// MI455X (gfx1250) — compile-verified
//
#include <hip/hip_runtime.h>
#include <hip/hip_bf16.h>

typedef float v2f __attribute__((ext_vector_type(2)));
typedef float v8f __attribute__((ext_vector_type(8)));

#define KVOX 12000
#define TPTS 35
#define FIN  7
#define C1H  16      // C1/2
#define C1F  32      // C1
#define C2H  64      // C2/2
#define C2F  128     // C2
#define CFD  128     // CF
#define GD   10
#define GH   400
#define GW   352
#define DHW  (GD*GH*GW)      // 1408000
#define NPTS (KVOX*TPTS)     // 420000
#define EPSV 1e-5f
#define MT   48              // T padded to 3 WMMA M-tiles

// ---------------- zero fill (float4 vectorized) ----------------
__global__ void k_fill_zero(float4* __restrict__ p, long long n4) {
    long long i = (long long)blockIdx.x * blockDim.x + threadIdx.x;
    long long stride = (long long)gridDim.x * blockDim.x;
    float4 z = make_float4(0.f, 0.f, 0.f, 0.f);
    for (; i < n4; i += stride) p[i] = z;
}

// ---------------- layer 1: 7 -> 16, relu, stats ----------------
__global__ void k_layer1(const float* __restrict__ x, const float* __restrict__ W1,
                         const float* __restrict__ b1, float* __restrict__ h1,
                         float* __restrict__ sum1, float* __restrict__ sq1) {
    __shared__ float s_s[C1H], s_q[C1H];
    int tid = threadIdx.x;
    if (tid < C1H) { s_s[tid] = 0.f; s_q[tid] = 0.f; }
    __syncthreads();
    int p = blockIdx.x * blockDim.x + tid;
    if (p < NPTS) {
        float xf[FIN];
        const float* xp = x + (size_t)p * FIN;
        #pragma unroll
        for (int f = 0; f < FIN; ++f) xf[f] = xp[f];
        #pragma unroll
        for (int c = 0; c < C1H; ++c) {
            float acc = b1[c];
            #pragma unroll
            for (int f = 0; f < FIN; ++f) acc += xf[f] * W1[f * C1H + c];
            acc = fmaxf(acc, 0.f);
            h1[(size_t)p * C1H + c] = acc;
            atomicAdd(&s_s[c], acc);
            atomicAdd(&s_q[c], acc * acc);
        }
    }
    __syncthreads();
    if (tid < C1H) {
        atomicAdd(&sum1[tid], s_s[tid]);
        atomicAdd(&sq1[tid], s_q[tid]);
    }
}

// ---------------- finalize BN stats ----------------
__global__ void k_stats_final(const float* __restrict__ sum, const float* __restrict__ sq,
                              float* __restrict__ mean, float* __restrict__ inv,
                              int C, float n) {
    int c = threadIdx.x;
    if (c < C) {
        float m = sum[c] / n;
        float v = sq[c] / n - m * m;
        mean[c] = m;
        inv[c] = rsqrtf(v + EPSV);
    }
}

// ---------------- build x2 tile (normalize h1, maxpool over T, concat, mask) ----------------
__device__ inline void build_x2(int k, int tid,
                                const float* __restrict__ x, const float* __restrict__ h1,
                                const float* __restrict__ mean1, const float* __restrict__ inv1,
                                const float* __restrict__ g1, const float* __restrict__ be1,
                                float (*s_x2)[33], float* s_mask, float* s_cmax1) {
    for (int idx = tid; idx < TPTS * C1H; idx += 256) {
        int t = idx >> 4, c = idx & 15;
        float hv = (h1[(size_t)(k * TPTS + t) * C1H + c] - mean1[c]) * inv1[c] * g1[c] + be1[c];
        s_x2[t][c] = hv;
    }
    for (int t = tid; t < MT; t += 256) {
        float mk = 0.f;
        if (t < TPTS) {
            const float* xp = x + (size_t)(k * TPTS + t) * FIN;
            float m = xp[0];
            #pragma unroll
            for (int f = 1; f < FIN; ++f) m = fmaxf(m, xp[f]);
            mk = (m != 0.f) ? 1.f : 0.f;
        }
        s_mask[t] = mk;
    }
    __syncthreads();
    if (tid < C1H) {
        float cm = -INFINITY;
        for (int t = 0; t < TPTS; ++t) cm = fmaxf(cm, s_x2[t][tid]);
        s_cmax1[tid] = cm;
    }
    __syncthreads();
    for (int idx = tid; idx < MT * C1F; idx += 256) {
        int t = idx >> 5, c = idx & 31;
        float v = 0.f;
        if (t < TPTS) {
            float hv = (c < C1H) ? s_x2[t][c] : s_cmax1[c - C1H];
            v = hv * s_mask[t];
        }
        s_x2[t][c] = v;
    }
    __syncthreads();
}

// ---------------- GEMM2 via f32 WMMA: (48x32) @ W2(32x64) + b2, relu -> s_pre ----------------
__device__ inline void gemm2_to_pre(int tid, const float* __restrict__ W2,
                                    const float* __restrict__ b2,
                                    float (*s_x2)[33], float (*s_pre)[66]) {
    int w = tid >> 5, lane = tid & 31, nl = lane & 15, kh = lane >> 4;
    if (w < 4) {
        int n0 = w * 16;
        float bias = b2[n0 + nl];
        v8f acc0, acc1, acc2;
        #pragma unroll
        for (int r = 0; r < 8; ++r) { acc0[r] = bias; acc1[r] = bias; acc2[r] = bias; }
        #pragma unroll
        for (int k0 = 0; k0 < C1F; k0 += 4) {
            v2f bfr;
            bfr.x = W2[(k0 + 2 * kh) * C2H + n0 + nl];
            bfr.y = W2[(k0 + 2 * kh + 1) * C2H + n0 + nl];
            v2f a0, a1, a2;
            a0.x = s_x2[nl][k0 + 2 * kh];      a0.y = s_x2[nl][k0 + 2 * kh + 1];
            a1.x = s_x2[16 + nl][k0 + 2 * kh]; a1.y = s_x2[16 + nl][k0 + 2 * kh + 1];
            a2.x = s_x2[32 + nl][k0 + 2 * kh]; a2.y = s_x2[32 + nl][k0 + 2 * kh + 1];
            acc0 = __builtin_amdgcn_wmma_f32_16x16x4_f32(false, a0, false, bfr, (short)0, acc0, false, false);
            acc1 = __builtin_amdgcn_wmma_f32_16x16x4_f32(false, a1, false, bfr, (short)0, acc1, false, false);
            acc2 = __builtin_amdgcn_wmma_f32_16x16x4_f32(false, a2, false, bfr, (short)0, acc2, false, false);
        }
        #pragma unroll
        for (int r = 0; r < 8; ++r) {
            s_pre[kh * 8 + r][n0 + nl]      = fmaxf(acc0[r], 0.f);
            s_pre[16 + kh * 8 + r][n0 + nl] = fmaxf(acc1[r], 0.f);
            s_pre[32 + kh * 8 + r][n0 + nl] = fmaxf(acc2[r], 0.f);
        }
    }
    __syncthreads();
}

// ---------------- layer2 GEMM for BN stats (h2pre discarded) ----------------
__global__ void __launch_bounds__(256)
k_layer2_stats(const float* __restrict__ x, const float* __restrict__ h1,
               const float* __restrict__ W2, const float* __restrict__ b2,
               const float* __restrict__ mean1, const float* __restrict__ inv1,
               const float* __restrict__ g1, const float* __restrict__ be1,
               float* __restrict__ sum2, float* __restrict__ sq2, int vpb) {
    __shared__ float s_x2[MT][33];
    __shared__ float s_pre[MT][66];
    __shared__ float s_mask[MT];
    __shared__ float s_cmax1[C1H];
    __shared__ float s_red[256];
    int tid = threadIdx.x;
    int c = tid & 63, q = tid >> 6;
    float lsum = 0.f, lsq = 0.f;
    for (int v = 0; v < vpb; ++v) {
        int k = blockIdx.x * vpb + v;
        if (k >= KVOX) break;
        build_x2(k, tid, x, h1, mean1, inv1, g1, be1, s_x2, s_mask, s_cmax1);
        gemm2_to_pre(tid, W2, b2, s_x2, s_pre);
        for (int t = q; t < TPTS; t += 4) {
            float val = s_pre[t][c];
            lsum += val;
            lsq += val * val;
        }
        __syncthreads();
    }
    s_red[tid] = lsum;
    __syncthreads();
    if (tid < 64) {
        float tot = s_red[tid] + s_red[tid + 64] + s_red[tid + 128] + s_red[tid + 192];
        atomicAdd(&sum2[tid], tot);
    }
    __syncthreads();
    s_red[tid] = lsq;
    __syncthreads();
    if (tid < 64) {
        float tot = s_red[tid] + s_red[tid + 64] + s_red[tid + 128] + s_red[tid + 192];
        atomicAdd(&sq2[tid], tot);
    }
}

// ---------------- fused final: rebuild x2, GEMM2, BN2, maxpool/concat, GEMM_f, T-max, scatter ----------------
__global__ void __launch_bounds__(256)
k_final(const float* __restrict__ x, const float* __restrict__ h1,
        const float* __restrict__ W2, const float* __restrict__ b2,
        const float* __restrict__ Wf, const float* __restrict__ bfv,
        const float* __restrict__ mean1, const float* __restrict__ inv1,
        const float* __restrict__ g1, const float* __restrict__ be1,
        const float* __restrict__ mean2, const float* __restrict__ inv2,
        const float* __restrict__ g2, const float* __restrict__ be2,
        const int* __restrict__ coor, float* __restrict__ out) {
    __shared__ float s_x2[MT][33];
    __shared__ float s_pre[MT][66];
    __shared__ float s_h2[MT][130];
    __shared__ float s_mask[MT];
    __shared__ float s_cmax1[C1H];
    __shared__ float s_cmax2[C2H];
    __shared__ float s_red[8][2][16];
    int tid = threadIdx.x;
    int k = blockIdx.x;

    // warm L2/L0 with Wf (64 KB, shared by all 12000 blocks) -> global_prefetch path
    {
        const char* wfb = (const char*)Wf;
        for (int off = tid * 128; off < C2F * CFD * 4; off += 256 * 128)
            __builtin_prefetch(wfb + off, 0, 2);
    }

    build_x2(k, tid, x, h1, mean1, inv1, g1, be1, s_x2, s_mask, s_cmax1);
    gemm2_to_pre(tid, W2, b2, s_x2, s_pre);

    // normalize h2pre -> hn2 (in place, valid rows only)
    for (int idx = tid; idx < TPTS * C2H; idx += 256) {
        int t = idx >> 6, c = idx & 63;
        float v = (s_pre[t][c] - mean2[c]) * inv2[c] * g2[c] + be2[c];
        s_pre[t][c] = v;
    }
    __syncthreads();
    if (tid < C2H) {
        float cm = -INFINITY;
        for (int t = 0; t < TPTS; ++t) cm = fmaxf(cm, s_pre[t][tid]);
        s_cmax2[tid] = cm;
    }
    __syncthreads();
    // build h2: [hn2 | broadcast maxpool] * mask, pad rows = 0
    for (int idx = tid; idx < MT * C2F; idx += 256) {
        int t = idx >> 7, c = idx & 127;
        float v = 0.f;
        if (t < TPTS) {
            float hv = (c < C2H) ? s_pre[t][c] : s_cmax2[c - C2H];
            v = hv * s_mask[t];
        }
        s_h2[t][c] = v;
    }
    __syncthreads();

    // final GEMM: (48x128) @ Wf(128x128) + bf ; k0 outer so each B-fragment
    // is loaded once and feeds all three M-tile WMMAs back-to-back
    int w = tid >> 5, lane = tid & 31, nl = lane & 15, kh = lane >> 4;
    int n0 = w * 16;
    float bias = bfv[n0 + nl];
    v8f acc0, acc1, acc2;
    #pragma unroll
    for (int r = 0; r < 8; ++r) { acc0[r] = bias; acc1[r] = bias; acc2[r] = bias; }
    for (int k0 = 0; k0 < C2F; k0 += 4) {
        v2f bfr;
        bfr.x = Wf[(k0 + 2 * kh) * CFD + n0 + nl];
        bfr.y = Wf[(k0 + 2 * kh + 1) * CFD + n0 + nl];
        v2f a0, a1, a2;
        a0.x = s_h2[nl][k0 + 2 * kh];      a0.y = s_h2[nl][k0 + 2 * kh + 1];
        a1.x = s_h2[16 + nl][k0 + 2 * kh]; a1.y = s_h2[16 + nl][k0 + 2 * kh + 1];
        a2.x = s_h2[32 + nl][k0 + 2 * kh]; a2.y = s_h2[32 + nl][k0 + 2 * kh + 1];
        acc0 = __builtin_amdgcn_wmma_f32_16x16x4_f32(false, a0, false, bfr, (short)0, acc0, false, false);
        acc1 = __builtin_amdgcn_wmma_f32_16x16x4_f32(false, a1, false, bfr, (short)0, acc1, false, false);
        acc2 = __builtin_amdgcn_wmma_f32_16x16x4_f32(false, a2, false, bfr, (short)0, acc2, false, false);
    }
    float pmax = -INFINITY;
    #pragma unroll
    for (int r = 0; r < 8; ++r) {
        int row0 = kh * 8 + r;       // M-tile 0 rows
        int row1 = 16 + kh * 8 + r;  // M-tile 1 rows
        int row2 = 32 + kh * 8 + r;  // M-tile 2 rows
        if (row0 < TPTS) pmax = fmaxf(pmax, acc0[r]);
        if (row1 < TPTS) pmax = fmaxf(pmax, acc1[r]);
        if (row2 < TPTS) pmax = fmaxf(pmax, acc2[r]);
    }
    s_red[w][kh][nl] = pmax;
    __syncthreads();

    if (tid < CFD) {
        int c = tid;
        float val = fmaxf(s_red[c >> 4][0][c & 15], s_red[c >> 4][1][c & 15]);
        int di = coor[k * 3 + 0];
        int hi = coor[k * 3 + 1];
        int wi = coor[k * 3 + 2];
        size_t lin = (size_t)di * (GH * GW) + (size_t)hi * GW + wi;
        out[(size_t)c * DHW + lin] = val;
    }
}

extern "C" void kernel_launch(void* const* d_in, const int* in_sizes, int n_in,
                              void* d_out, int out_size, void* d_ws, size_t ws_size,
                              hipStream_t stream) {
    const float* x    = (const float*)d_in[0];
    const int*   coor = (const int*)d_in[1];
    // d_in[2] = shape tuple (hardcoded)
    const float* W1  = (const float*)d_in[3];
    const float* b1  = (const float*)d_in[4];
    const float* g1  = (const float*)d_in[5];
    const float* be1 = (const float*)d_in[6];
    const float* W2  = (const float*)d_in[7];
    const float* b2  = (const float*)d_in[8];
    const float* g2  = (const float*)d_in[9];
    const float* be2 = (const float*)d_in[10];
    const float* Wf  = (const float*)d_in[11];
    const float* bf  = (const float*)d_in[12];

    float* ws    = (float*)d_ws;
    float* sum1  = ws + 0;
    float* sq1   = ws + 16;
    float* mean1 = ws + 32;
    float* inv1  = ws + 48;
    float* sum2  = ws + 64;
    float* sq2   = ws + 128;
    float* mean2 = ws + 192;
    float* inv2  = ws + 256;
    float* h1    = ws + 512;
    float* outp  = (float*)d_out;

    // 1) zero the 721 MB dense output (memory-floor of the whole workload)
    long long n4 = (long long)out_size / 4;
    k_fill_zero<<<4096, 256, 0, stream>>>((float4*)outp, n4);
    // zero stats accumulators (512 floats = 128 float4)
    k_fill_zero<<<1, 128, 0, stream>>>((float4*)d_ws, 128);

    // 2) layer1 + stats1
    k_layer1<<<(NPTS + 255) / 256, 256, 0, stream>>>(x, W1, b1, h1, sum1, sq1);
    k_stats_final<<<1, 64, 0, stream>>>(sum1, sq1, mean1, inv1, C1H, (float)NPTS);

    // 3) layer2 GEMM (WMMA) for stats2 only
    k_layer2_stats<<<KVOX / 8, 256, 0, stream>>>(x, h1, W2, b2, mean1, inv1, g1, be1,
                                                 sum2, sq2, 8);
    k_stats_final<<<1, 64, 0, stream>>>(sum2, sq2, mean2, inv2, C2H, (float)NPTS);

    // 4) fused final per-voxel kernel (WMMA GEMM2 + WMMA 128x128 GEMM + scatter)
    k_final<<<KVOX, 256, 0, stream>>>(x, h1, W2, b2, Wf, bf,
                                      mean1, inv1, g1, be1,
                                      mean2, inv2, g2, be2,
                                      coor, outp);
}